// NodeModel_52415780880560
// MI455X (gfx1250) — compile-verified
//
#include <hip/hip_runtime.h>

typedef float v2f __attribute__((ext_vector_type(2)));
typedef float v8f __attribute__((ext_vector_type(8)));

#define AP 68          // A-tile row stride (floats), padded for LDS bank spread
#define WP2 80         // weight pair-row stride (float2 units); 160 dwords == 32 mod 64 banks
#define WAVES 4
#define BLK 128        // 4 wave32 per block

__device__ __forceinline__ v8f zero8() {
  v8f z = {0.f,0.f,0.f,0.f,0.f,0.f,0.f,0.f};
  return z;
}

__device__ __forceinline__ v8f wmma4(v2f a, v2f b, v8f c) {
  // D = A(16x4,f32) * B(4x16,f32) + C(16x16,f32), exact fp32 matrix op
  return __builtin_amdgcn_wmma_f32_16x16x4_f32(false, a, false, b, (short)0, c,
                                               false, false);
}

// Logical padded GEMM1 weight row k (0..67):
//   k<64 -> w[k+3] (feature part), k<67 -> w[k-64] (x part), k==67 -> 0
__device__ __forceinline__ float w1row(const float* __restrict__ w, int k, int j) {
  if (k < 64) return w[(k + 3) * 64 + j];
  if (k < 67) return w[(k - 64) * 64 + j];
  return 0.0f;
}

// Pack GEMM1 weights as K-pairs: wt[kp*WP2 + j] = {row(2kp)[j], row(2kp+1)[j]}
__device__ __forceinline__ void load_w_gemm1(v2f* wt, const float* __restrict__ w) {
  for (int idx = threadIdx.x; idx < 34 * 64; idx += BLK) {
    int kp = idx >> 6, j = idx & 63;
    v2f p;
    p.x = w1row(w, 2 * kp, j);
    p.y = w1row(w, 2 * kp + 1, j);
    wt[kp * WP2 + j] = p;
  }
}

__device__ __forceinline__ void load_w_gemm2(v2f* wt, const float* __restrict__ w) {
  for (int idx = threadIdx.x; idx < 32 * 64; idx += BLK) {
    int kp = idx >> 6, j = idx & 63;
    v2f p;
    p.x = w[(2 * kp) * 64 + j];
    p.y = w[(2 * kp + 1) * 64 + j];
    wt[kp * WP2 + j] = p;
  }
}

// c[nt] (nt = N-tile of 16) += A(16 x 4*KSTEPS) * W(4*KSTEPS x 64), fully unrolled
template <int KSTEPS>
__device__ __forceinline__ void gemm_tile(const float* at, const v2f* wt,
                                          int nl, int kh, v8f c[4]) {
#pragma unroll
  for (int ki = 0; ki < KSTEPS; ++ki) {
    int k0 = 4 * ki + 2 * kh;
    v2f a = *(const v2f*)(at + nl * AP + k0);   // ds_load_b64, pair-aligned
#pragma unroll
    for (int nt = 0; nt < 4; ++nt) {
      v2f b = wt[(2 * ki + kh) * WP2 + nt * 16 + nl];  // ds_load_b64, pre-paired
      c[nt] = wmma4(a, b, c[nt]);
    }
  }
}

// ---------------- Pass 1: edge GEMM1 + BN statistics ----------------
__global__ void __launch_bounds__(BLK)
edge_stats_kernel(const float* __restrict__ x, const int* __restrict__ eidx,
                  const float* __restrict__ eattr, const float* __restrict__ w1,
                  const float* __restrict__ b1, float* __restrict__ gstats,
                  long E, long NT) {
  __shared__ v2f wt[34 * WP2];
  __shared__ float atile[WAVES * 16 * AP];
  __shared__ float sdata[128];

  load_w_gemm1(wt, w1);
  if (threadIdx.x < 128) sdata[threadIdx.x] = 0.0f;
  __syncthreads();

  const int wid = threadIdx.x >> 5;
  const int lane = threadIdx.x & 31;
  const int nl = lane & 15, kh = lane >> 4;
  float* at = &atile[wid * 16 * AP];

  float bias[4];
#pragma unroll
  for (int nt = 0; nt < 4; ++nt) bias[nt] = b1[nt * 16 + nl];

  float sumv[4] = {0.f, 0.f, 0.f, 0.f}, sqv[4] = {0.f, 0.f, 0.f, 0.f};

  for (long t = (long)blockIdx.x * WAVES + wid; t < NT; t += (long)gridDim.x * WAVES) {
    long e0 = t * 16;
    for (int r = 0; r < 16; ++r) {
      long e = e0 + r;
      const float* ap = eattr + e * 64;
      float2 v = *(const float2*)(ap + 2 * lane);
      at[r * AP + 2 * lane] = v.x;
      at[r * AP + 2 * lane + 1] = v.y;
      if (lane < 3) {
        int src = eidx[e];
        at[r * AP + 64 + lane] = x[(long)src * 3 + lane];
      } else if (lane == 3) {
        at[r * AP + 67] = 0.0f;
      }
    }
    v8f c[4] = {zero8(), zero8(), zero8(), zero8()};
    gemm_tile<17>(at, wt, nl, kh, c);
#pragma unroll
    for (int nt = 0; nt < 4; ++nt)
#pragma unroll
      for (int r = 0; r < 8; ++r) {
        float v = c[nt][r] + bias[nt];
        sumv[nt] += v;
        sqv[nt] += v * v;
      }
  }
#pragma unroll
  for (int nt = 0; nt < 4; ++nt) {
    sumv[nt] += __shfl_xor(sumv[nt], 16, 32);
    sqv[nt]  += __shfl_xor(sqv[nt], 16, 32);
  }
  if (kh == 0) {
#pragma unroll
    for (int nt = 0; nt < 4; ++nt) {
      atomicAdd(&sdata[nt * 16 + nl], sumv[nt]);
      atomicAdd(&sdata[64 + nt * 16 + nl], sqv[nt]);
    }
  }
  __syncthreads();
  if (threadIdx.x < 128) unsafeAtomicAdd(&gstats[threadIdx.x], sdata[threadIdx.x]);
}

// ---------------- BN finalize: scale/shift ----------------
__global__ void finalize_kernel(const float* __restrict__ gstats,
                                const float* __restrict__ gamma,
                                const float* __restrict__ beta, float invCount,
                                float* __restrict__ scale, float* __restrict__ shift) {
  int j = threadIdx.x;
  float mean = gstats[j] * invCount;
  float var = fmaxf(gstats[64 + j] * invCount - mean * mean, 0.0f);
  float sc = gamma[j] * rsqrtf(var + 1e-5f);
  scale[j] = sc;
  shift[j] = beta[j] - mean * sc;
}

// ---------------- Pass 2: edge MLP + scatter-add ----------------
__global__ void __launch_bounds__(BLK)
edge_scatter_kernel(const float* __restrict__ x, const int* __restrict__ eidx,
                    const float* __restrict__ eattr, const float* __restrict__ w1,
                    const float* __restrict__ b1, const float* __restrict__ w2,
                    const float* __restrict__ b2, const float* __restrict__ sc1,
                    const float* __restrict__ sh1, float* __restrict__ s,
                    float* __restrict__ cnt, long E, long NT) {
  __shared__ v2f wt1[34 * WP2];
  __shared__ v2f wt2[32 * WP2];
  __shared__ float atile[WAVES * 16 * AP];
  __shared__ int colbuf[WAVES * 16];

  load_w_gemm1(wt1, w1);
  load_w_gemm2(wt2, w2);
  __syncthreads();

  const int wid = threadIdx.x >> 5;
  const int lane = threadIdx.x & 31;
  const int nl = lane & 15, kh = lane >> 4;
  float* at = &atile[wid * 16 * AP];
  int* cb = &colbuf[wid * 16];

  float sc[4], shf[4], bias2v[4];
#pragma unroll
  for (int nt = 0; nt < 4; ++nt) {
    int n = nt * 16 + nl;
    float scv = sc1[n];
    sc[nt] = scv;
    shf[nt] = fmaf(b1[n], scv, sh1[n]);  // fold linear bias into BN shift
    bias2v[nt] = b2[n];
  }

  for (long t = (long)blockIdx.x * WAVES + wid; t < NT; t += (long)gridDim.x * WAVES) {
    long e0 = t * 16;
    for (int r = 0; r < 16; ++r) {
      long e = e0 + r;
      const float* ap = eattr + e * 64;
      float2 v = *(const float2*)(ap + 2 * lane);
      at[r * AP + 2 * lane] = v.x;
      at[r * AP + 2 * lane + 1] = v.y;
      if (lane < 3) {
        int src = eidx[e];
        at[r * AP + 64 + lane] = x[(long)src * 3 + lane];
      } else if (lane == 3) {
        at[r * AP + 67] = 0.0f;
      } else if (lane == 4) {
        cb[r] = eidx[E + e];  // destination node
      }
    }
    v8f c[4] = {zero8(), zero8(), zero8(), zero8()};
    gemm_tile<17>(at, wt1, nl, kh, c);
    // BN + ReLU, transpose C-layout -> A-layout through the (now free) LDS tile
#pragma unroll
    for (int nt = 0; nt < 4; ++nt)
#pragma unroll
      for (int r = 0; r < 8; ++r)
        at[(r + 8 * kh) * AP + nt * 16 + nl] =
            fmaxf(fmaf(c[nt][r], sc[nt], shf[nt]), 0.0f);
    v8f d[4] = {zero8(), zero8(), zero8(), zero8()};
    gemm_tile<16>(at, wt2, nl, kh, d);
    // scatter-add into s[col], f32 hardware atomics (L2-resident 25.6MB)
#pragma unroll
    for (int nt = 0; nt < 4; ++nt)
#pragma unroll
      for (int r = 0; r < 8; ++r) {
        int cidx = cb[r + 8 * kh];
        unsafeAtomicAdd(&s[(long)cidx * 64 + nt * 16 + nl], d[nt][r] + bias2v[nt]);
      }
    if (lane < 16) unsafeAtomicAdd(&cnt[cb[lane]], 1.0f);
  }
}

// ---------------- Pass 3: node GEMM1 + BN statistics ----------------
__global__ void __launch_bounds__(BLK)
node_stats_kernel(const float* __restrict__ x, const float* __restrict__ s,
                  const float* __restrict__ cnt, const float* __restrict__ w3,
                  const float* __restrict__ b3, float* __restrict__ gstats, long NT) {
  __shared__ v2f wt[34 * WP2];
  __shared__ float atile[WAVES * 16 * AP];
  __shared__ float sdata[128];

  load_w_gemm1(wt, w3);
  if (threadIdx.x < 128) sdata[threadIdx.x] = 0.0f;
  __syncthreads();

  const int wid = threadIdx.x >> 5;
  const int lane = threadIdx.x & 31;
  const int nl = lane & 15, kh = lane >> 4;
  float* at = &atile[wid * 16 * AP];

  float bias[4];
#pragma unroll
  for (int nt = 0; nt < 4; ++nt) bias[nt] = b3[nt * 16 + nl];

  float sumv[4] = {0.f, 0.f, 0.f, 0.f}, sqv[4] = {0.f, 0.f, 0.f, 0.f};

  for (long t = (long)blockIdx.x * WAVES + wid; t < NT; t += (long)gridDim.x * WAVES) {
    long i0 = t * 16;
    for (int r = 0; r < 16; ++r) {
      long i = i0 + r;
      float inv = 1.0f / fmaxf(cnt[i], 1.0f);
      float2 v = *(const float2*)(s + i * 64 + 2 * lane);
      at[r * AP + 2 * lane] = v.x * inv;
      at[r * AP + 2 * lane + 1] = v.y * inv;
      if (lane < 3)       at[r * AP + 64 + lane] = x[i * 3 + lane];
      else if (lane == 3) at[r * AP + 67] = 0.0f;
    }
    v8f c[4] = {zero8(), zero8(), zero8(), zero8()};
    gemm_tile<17>(at, wt, nl, kh, c);
#pragma unroll
    for (int nt = 0; nt < 4; ++nt)
#pragma unroll
      for (int r = 0; r < 8; ++r) {
        float v = c[nt][r] + bias[nt];
        sumv[nt] += v;
        sqv[nt] += v * v;
      }
  }
#pragma unroll
  for (int nt = 0; nt < 4; ++nt) {
    sumv[nt] += __shfl_xor(sumv[nt], 16, 32);
    sqv[nt]  += __shfl_xor(sqv[nt], 16, 32);
  }
  if (kh == 0) {
#pragma unroll
    for (int nt = 0; nt < 4; ++nt) {
      atomicAdd(&sdata[nt * 16 + nl], sumv[nt]);
      atomicAdd(&sdata[64 + nt * 16 + nl], sqv[nt]);
    }
  }
  __syncthreads();
  if (threadIdx.x < 128) unsafeAtomicAdd(&gstats[threadIdx.x], sdata[threadIdx.x]);
}

// ---------------- Pass 4: node MLP -> output ----------------
__global__ void __launch_bounds__(BLK)
node_out_kernel(const float* __restrict__ x, const float* __restrict__ s,
                const float* __restrict__ cnt, const float* __restrict__ w3,
                const float* __restrict__ b3, const float* __restrict__ w4,
                const float* __restrict__ b4, const float* __restrict__ sc3,
                const float* __restrict__ sh3, float* __restrict__ out, long NT) {
  __shared__ v2f wt3[34 * WP2];
  __shared__ v2f wt4[32 * WP2];
  __shared__ float atile[WAVES * 16 * AP];

  load_w_gemm1(wt3, w3);
  load_w_gemm2(wt4, w4);
  __syncthreads();

  const int wid = threadIdx.x >> 5;
  const int lane = threadIdx.x & 31;
  const int nl = lane & 15, kh = lane >> 4;
  float* at = &atile[wid * 16 * AP];

  float sc[4], shf[4], bias4v[4];
#pragma unroll
  for (int nt = 0; nt < 4; ++nt) {
    int n = nt * 16 + nl;
    float scv = sc3[n];
    sc[nt] = scv;
    shf[nt] = fmaf(b3[n], scv, sh3[n]);
    bias4v[nt] = b4[n];
  }

  for (long t = (long)blockIdx.x * WAVES + wid; t < NT; t += (long)gridDim.x * WAVES) {
    long i0 = t * 16;
    for (int r = 0; r < 16; ++r) {
      long i = i0 + r;
      float inv = 1.0f / fmaxf(cnt[i], 1.0f);
      float2 v = *(const float2*)(s + i * 64 + 2 * lane);
      at[r * AP + 2 * lane] = v.x * inv;
      at[r * AP + 2 * lane + 1] = v.y * inv;
      if (lane < 3)       at[r * AP + 64 + lane] = x[i * 3 + lane];
      else if (lane == 3) at[r * AP + 67] = 0.0f;
    }
    v8f c[4] = {zero8(), zero8(), zero8(), zero8()};
    gemm_tile<17>(at, wt3, nl, kh, c);
#pragma unroll
    for (int nt = 0; nt < 4; ++nt)
#pragma unroll
      for (int r = 0; r < 8; ++r)
        at[(r + 8 * kh) * AP + nt * 16 + nl] =
            fmaxf(fmaf(c[nt][r], sc[nt], shf[nt]), 0.0f);
    v8f d[4] = {zero8(), zero8(), zero8(), zero8()};
    gemm_tile<16>(at, wt4, nl, kh, d);
#pragma unroll
    for (int nt = 0; nt < 4; ++nt)
#pragma unroll
      for (int r = 0; r < 8; ++r)
        out[(i0 + r + 8 * kh) * 64 + nt * 16 + nl] = d[nt][r] + bias4v[nt];
  }
}

extern "C" void kernel_launch(void* const* d_in, const int* in_sizes, int n_in,
                              void* d_out, int out_size, void* d_ws, size_t ws_size,
                              hipStream_t stream) {
  const float* x     = (const float*)d_in[0];
  const int*   eidx  = (const int*)d_in[1];     // [2,E] int32: row, then col
  const float* eattr = (const float*)d_in[2];
  const float* w1 = (const float*)d_in[5];
  const float* b1 = (const float*)d_in[6];
  const float* g1 = (const float*)d_in[7];
  const float* be1 = (const float*)d_in[8];
  const float* w2 = (const float*)d_in[9];
  const float* b2 = (const float*)d_in[10];
  const float* w3 = (const float*)d_in[11];
  const float* b3 = (const float*)d_in[12];
  const float* g3 = (const float*)d_in[13];
  const float* be3 = (const float*)d_in[14];
  const float* w4 = (const float*)d_in[15];
  const float* b4 = (const float*)d_in[16];

  long N = (long)in_sizes[0] / 3;
  long E = (long)in_sizes[1] / 2;
  float* out = (float*)d_out;

  // workspace layout (floats): s[N*64] | cnt[N] | gst1[128] | gst3[128] |
  //                            sc1[64] | sh1[64] | sc3[64] | sh3[64]
  float* wsf  = (float*)d_ws;
  float* sbuf = wsf;
  float* cnt  = sbuf + N * 64;
  float* gst1 = cnt + N;
  float* gst3 = gst1 + 128;
  float* sc1  = gst3 + 128;
  float* sh1  = sc1 + 64;
  float* sc3  = sh1 + 64;
  float* sh3  = sc3 + 64;

  // zero the accumulated regions every launch (s, cnt, gst1, gst3 contiguous)
  hipMemsetAsync(sbuf, 0, (size_t)(N * 64 + N + 256) * sizeof(float), stream);

  long NTe = E / 16;  // 100000 (E divisible by 16)
  long NTn = N / 16;  // 6250   (N divisible by 16)

  int gridE = (int)((NTe + WAVES - 1) / WAVES);
  if (gridE > 2048) gridE = 2048;
  int gridN = (int)((NTn + WAVES - 1) / WAVES);
  if (gridN > 1024) gridN = 1024;

  edge_stats_kernel<<<gridE, BLK, 0, stream>>>(x, eidx, eattr, w1, b1, gst1, E, NTe);
  finalize_kernel<<<1, 64, 0, stream>>>(gst1, g1, be1, 1.0f / (float)E, sc1, sh1);
  edge_scatter_kernel<<<gridE, BLK, 0, stream>>>(x, eidx, eattr, w1, b1, w2, b2,
                                                 sc1, sh1, sbuf, cnt, E, NTe);
  node_stats_kernel<<<gridN, BLK, 0, stream>>>(x, sbuf, cnt, w3, b3, gst3, NTn);
  finalize_kernel<<<1, 64, 0, stream>>>(gst3, g3, be3, 1.0f / (float)N, sc3, sh3);
  node_out_kernel<<<gridN, BLK, 0, stream>>>(x, sbuf, cnt, w3, b3, w4, b4,
                                             sc3, sh3, out, NTn);
}